// CenterLayer_63840393887803
// MI455X (gfx1250) — compile-verified
//
#include <hip/hip_runtime.h>

// ---------------------------------------------------------------------------
// Center-loss layer for MI455X (gfx1250, wave32).
//
//   loss[b]        = sum_f (centers[y_true[b]][f] - y_pred[b][f])^2
//   delta[c][f]    = sum_{b: y_true[b]==c} diff[b][f]         (scatter-add)
//   counts[c]      = |{b: y_true[b]==c}|
//   new_centers    = centers - 0.5 * delta / (counts + 1)
//
// Memory-bound: ~75 MB HBM traffic (~3.2us floor @ 23.3 TB/s); scatter adds
// (16.7M f32 atomics) stay in the L2-resident 10MB delta array. The per-row
// loss reduction rides the matrix pipe via V_WMMA_F32_16X16X4_F32 with
// A = ones(16x4), B[k,n] = diff^2 (column n = batch row n), accumulating
// D[m,n] = loss[row n]; VALU only produces diff / diff^2.
// ---------------------------------------------------------------------------

typedef __attribute__((ext_vector_type(2))) float v2f;
typedef __attribute__((ext_vector_type(4))) float v4f;
typedef __attribute__((ext_vector_type(8))) float v8f;

#define NUM_CLASSES 10000
#define FEAT        256
#define BATCH       65536
#define ALPHA       0.5f

__device__ __forceinline__ void atomic_add_f32(float* p, float v) {
  // Relaxed agent-scope FP add -> global_atomic_add_f32 (no return, STOREcnt).
  (void)__hip_atomic_fetch_add(p, v, __ATOMIC_RELAXED, __HIP_MEMORY_SCOPE_AGENT);
}

// ---- kernel 1: zero delta[C][F] + counts[C] (float4 granularity) ----------
__global__ __launch_bounds__(256) void zero_ws_kernel(float* __restrict__ ws,
                                                      int n4) {
  int i = blockIdx.x * blockDim.x + threadIdx.x;
  if (i < n4) {
    v4f z = {0.f, 0.f, 0.f, 0.f};
    *(v4f*)(ws + (size_t)i * 4) = z;
  }
}

// ---- kernel 2: gather + diff + scatter-atomics + WMMA loss reduction ------
// Block = 256 threads = 8 waves; each wave owns 16 batch rows.
// Lane pair (n, n+16) co-owns row r0+n:
//   half=0 lane supplies B-matrix K=0,1  -> features 4j+0, 4j+1
//   half=1 lane supplies B-matrix K=2,3  -> features 4j+2, 4j+3
// matching the 32-bit B-operand VGPR layout of V_WMMA_F32_16X16X4_F32.
__global__ __launch_bounds__(256) void center_main_kernel(
    const float* __restrict__ y_pred,   // [BATCH, FEAT]
    const float* __restrict__ centers,  // [NUM_CLASSES, FEAT]
    const int*   __restrict__ y_true,   // [BATCH]
    float*       __restrict__ loss,     // [BATCH]
    float*       __restrict__ delta,    // [NUM_CLASSES, FEAT] (ws, zeroed)
    float*       __restrict__ counts)   // [NUM_CLASSES]       (ws, zeroed)
{
  const int lane = threadIdx.x & 31;
  const int wave = threadIdx.x >> 5;
  const int half = lane >> 4;   // which K-pair this lane feeds into B
  const int n    = lane & 15;   // batch row within the 16-row tile
  const int row  = (blockIdx.x * 8 + wave) * 16 + n;

  const int cls = y_true[row];
  const float* yp = y_pred  + (size_t)row * FEAT;
  const float* ce = centers + (size_t)cls * FEAT;
  float*       dl = delta   + (size_t)cls * FEAT;

  v8f acc = {0.f, 0.f, 0.f, 0.f, 0.f, 0.f, 0.f, 0.f};
  const v2f ones = {1.f, 1.f};  // A = 16x4 all-ones (uniform across lanes)

  // 256 features = 64 K-chunks of 4; no divergence -> EXEC all-ones for WMMA.
#pragma unroll 4
  for (int j = 0; j < 64; ++j) {
    const int f = 4 * j + 2 * half;
    v2f p = *(const v2f*)(yp + f);   // global_load_b64
    v2f c = *(const v2f*)(ce + f);   // gather, L2-resident
    v2f d = c - p;

    atomic_add_f32(dl + f,     d.x); // scatter-add into class delta
    atomic_add_f32(dl + f + 1, d.y);

    v2f sq = d * d;                  // B operand = diff^2
    // D = A(ones) x B(diff^2) + C  ==> every D row accumulates per-column
    // (= per-batch-row) sums of diff^2.
    acc = __builtin_amdgcn_wmma_f32_16x16x4_f32(
        /*neg_a=*/false, ones, /*neg_b=*/false, sq,
        /*c_mod=*/(short)0, acc, /*reuse_a=*/false, /*reuse_b=*/false);
  }

  if (half == 0) {
    // D layout: lane n (0..15), VGPR0 holds D[M=0, N=n] = loss of row r0+n.
    loss[row] = acc[0];
    atomic_add_f32(counts + cls, 1.0f);
  }
}

// ---- kernel 3: new_centers = centers - ALPHA * delta / (counts + 1) -------
__global__ __launch_bounds__(256) void finalize_kernel(
    const float* __restrict__ centers,
    const float* __restrict__ delta,
    const float* __restrict__ counts,
    float*       __restrict__ out)    // [NUM_CLASSES, FEAT]
{
  int i = blockIdx.x * blockDim.x + threadIdx.x;  // one float4 per thread
  int c = i >> 6;                                 // 64 float4 per class row
  v4f ce = *(const v4f*)(centers + (size_t)i * 4);
  v4f dl = *(const v4f*)(delta   + (size_t)i * 4);
  float s = ALPHA / (counts[c] + 1.0f);
  v4f r = ce - dl * s;
  *(v4f*)(out + (size_t)i * 4) = r;
}

extern "C" void kernel_launch(void* const* d_in, const int* in_sizes, int n_in,
                              void* d_out, int out_size, void* d_ws,
                              size_t ws_size, hipStream_t stream) {
  (void)in_sizes; (void)n_in; (void)out_size; (void)ws_size;

  const float* y_pred  = (const float*)d_in[0];
  const float* centers = (const float*)d_in[1];
  const int*   y_true  = (const int*)d_in[2];

  float* loss        = (float*)d_out;           // [BATCH]
  float* new_centers = (float*)d_out + BATCH;   // [NUM_CLASSES*FEAT]

  float* delta  = (float*)d_ws;                     // 10000*256 floats
  float* counts = delta + NUM_CLASSES * FEAT;       // 10000 floats

  // 1) zero workspace (delta + counts), float4 granularity
  const int n4 = (NUM_CLASSES * FEAT + NUM_CLASSES) / 4;  // 642,500
  zero_ws_kernel<<<(n4 + 255) / 256, 256, 0, stream>>>((float*)d_ws, n4);

  // 2) fused gather/diff/scatter/loss: 512 blocks x 8 waves x 16 rows = 65536
  center_main_kernel<<<BATCH / 128, 256, 0, stream>>>(
      y_pred, centers, y_true, loss, delta, counts);

  // 3) finalize new_centers: 640,000 float4 elements
  finalize_kernel<<<(NUM_CLASSES * FEAT / 4) / 256, 256, 0, stream>>>(
      centers, delta, counts, new_centers);
}